// ClusterNCALoss_49503793054560
// MI455X (gfx1250) — compile-verified
//
#include <hip/hip_runtime.h>
#include <math.h>

#define NS   65536
#define DIMF 128
#define NC   1024
#define ALPHA_F 16.0f
#define EPS_F   1e-3f

typedef __attribute__((ext_vector_type(2))) float v2f;
typedef __attribute__((ext_vector_type(8))) float v8f;
typedef unsigned int u32x4 __attribute__((ext_vector_type(4)));
typedef int i32x4 __attribute__((ext_vector_type(4)));
typedef int i32x8 __attribute__((ext_vector_type(8)));

// ---------------- init: zero accumulators + output ----------------
__global__ __launch_bounds__(256)
void k_init(float* __restrict__ centers, float* __restrict__ sizes,
            int* __restrict__ ctg, float* __restrict__ out) {
    int idx = blockIdx.x * blockDim.x + threadIdx.x;
    if (idx < NC * DIMF) centers[idx] = 0.0f;
    if (idx < NC) { sizes[idx] = 0.0f; ctg[idx] = 0; }
    if (idx == 0) out[0] = 0.0f;
}

// ---------------- segment-sum of inputs into center sums ----------------
__global__ __launch_bounds__(256)
void k_accum(const float* __restrict__ x, const int* __restrict__ cid,
             const int* __restrict__ tgt, float* __restrict__ centers,
             float* __restrict__ sizes, int* __restrict__ ctg) {
    int idx = blockIdx.x * blockDim.x + threadIdx.x;   // over NS*DIMF
    int i = idx >> 7;
    int k = idx & (DIMF - 1);
    int c = cid[i];
    unsafeAtomicAdd(&centers[c * DIMF + k], x[idx]);
    if (k == 0) {
        unsafeAtomicAdd(&sizes[c], 1.0f);
        ctg[c] = tgt[i];   // clusters are class-pure: any member's label works
    }
}

// ---------------- per-row squared norms ----------------
__global__ __launch_bounds__(256)
void k_rownorm(const float* __restrict__ x, float* __restrict__ xx) {
    int i = blockIdx.x * blockDim.x + threadIdx.x;     // over NS
    const float4* p = (const float4*)(x + (size_t)i * DIMF);
    float s = 0.0f;
    #pragma unroll
    for (int t = 0; t < DIMF / 4; ++t) {
        float4 v = p[t];
        s += v.x * v.x + v.y * v.y + v.z * v.z + v.w * v.w;
    }
    xx[i] = s;
}

// ---------------- finalize centers: mean, ||c||^2, size weight ----------------
__global__ __launch_bounds__(128)
void k_fin(float* __restrict__ centers, const float* __restrict__ sizes,
           float* __restrict__ yy, float* __restrict__ w) {
    int c    = blockIdx.x * 4 + (threadIdx.x >> 5);    // one wave32 per cluster
    int lane = threadIdx.x & 31;
    float s   = sizes[c];
    float inv = 1.0f / s;
    float4* cp = (float4*)(centers + (size_t)c * DIMF);
    float4 v = cp[lane];
    v.x *= inv; v.y *= inv; v.z *= inv; v.w *= inv;
    cp[lane] = v;
    float sq = v.x * v.x + v.y * v.y + v.z * v.z + v.w * v.w;
    #pragma unroll
    for (int m = 16; m >= 1; m >>= 1) sq += __shfl_xor(sq, m, 32);
    if (lane == 0) {
        yy[c] = sq;
        float r = s / (s - 1.0f);
        w[c] = (s > 1.0f) ? r * r : 1.0f;
    }
}

// ---------------- TDM helpers ----------------
typedef __attribute__((address_space(3))) const char lds_char;

__device__ __forceinline__ unsigned lds_offset(const void* p) {
    return (unsigned)(unsigned long long)(lds_char*)p;   // LDS byte offset (AS3)
}

// Issue a TDM load of a 16x128 f32 tile (row stride 128 elements) into LDS.
// D# built per CDNA5 ISA 8.3/8.4: group0 = {count, lds_addr, global_addr, type=2},
// group1 = {data_size=4B, tensor_dim0=128, tensor_dim1=16, tile_dim0=128,
//           tile_dim1=16, tensor_dim0_stride=128}. Remaining groups zero (2D tile).
__device__ __forceinline__ void tdm_load_tile16x128(const float* gsrc, unsigned ldsoff) {
    unsigned long long ga = (unsigned long long)(uintptr_t)gsrc;
    u32x4 g0;
    g0.x = 1u;                                               // count=1, user desc
    g0.y = ldsoff;                                           // lds_addr (bytes)
    g0.z = (unsigned)(ga & 0xFFFFFFFFu);                     // global_addr[31:0]
    g0.w = (unsigned)((ga >> 32) & 0x01FFFFFFu) | (2u << 30);// addr[56:32] | type=2
    i32x8 g1;
    g1.s0 = (int)(2u << 16);          // data_size=2 (4 bytes), workgroup_mask=0
    g1.s1 = (int)(128u << 16);        // tensor_dim0[15:0]=128 @ bits[63:48]
    g1.s2 = (int)(16u  << 16);        // tensor_dim1[15:0]=16  @ bits[95:80]
    g1.s3 = (int)(128u << 16);        // tile_dim0=128         @ bits[127:112]
    g1.s4 = 16;                       // tile_dim1=16; tile_dim2=0
    g1.s5 = 128;                      // tensor_dim0_stride[31:0]=128 elements
    g1.s6 = 0;
    g1.s7 = 0;
    i32x4 g2 = {0, 0, 0, 0};
    i32x4 g3 = {0, 0, 0, 0};
    i32x8 g4 = {0, 0, 0, 0, 0, 0, 0, 0};
    __builtin_amdgcn_tensor_load_to_lds(g0, g1, g2, g3, g4, 0);
}

// ---------------- main: TDM-fed WMMA distance tiles + NCA loss ----------------
// 4 waves per block; each wave owns a 16-row tile (block covers 64 rows).
// Center tiles are TDM-staged into double-buffered LDS and shared by all waves.
// Pass 1 finds masked max/min (-> base); pass 2 recomputes D via WMMA (cheaper
// than 64KB/wave LDS spill) and accumulates the masked exp sums.
__global__ __launch_bounds__(128)
void k_main(const float* __restrict__ x, const int* __restrict__ cid,
            const int* __restrict__ tgt, const float* __restrict__ centers,
            const float* __restrict__ yy, const float* __restrict__ w,
            const float* __restrict__ xx, const int* __restrict__ ctg,
            float* __restrict__ out) {
    __shared__ float sB[2][16 * DIMF];             // 2 x 8KB center tiles

    const int lane = threadIdx.x & 31;             // wave32
    const int wv   = threadIdx.x >> 5;             // wave id 0..3
    const int lo   = lane & 15;
    const int hi   = lane >> 4;
    const int row_base = blockIdx.x * 64 + wv * 16;

    // Preload A tile (16 rows x 128) in WMMA 32-bit A layout:
    // lane<16 holds (K=4kk,4kk+1) of row M=lo; lane>=16 holds (K=4kk+2,4kk+3).
    v2f av[32];
    const float* aptr = x + (size_t)(row_base + lo) * DIMF + 2 * hi;
    #pragma unroll
    for (int kk = 0; kk < 32; ++kk) av[kk] = *(const v2f*)(aptr + kk * 4);

    // Metadata for the 8 rows this lane owns in the C/D layout (M = j + 8*hi).
    int   rcid[8], rtgt[8];
    float rxx[8];
    #pragma unroll
    for (int j = 0; j < 8; ++j) {
        int r = row_base + j + 8 * hi;
        rcid[j] = cid[r];
        rtgt[j] = tgt[r];
        rxx[j]  = xx[r];
    }

    float dmax[8], dmin[8];
    #pragma unroll
    for (int j = 0; j < 8; ++j) { dmax[j] = -__builtin_inff(); dmin[j] = __builtin_inff(); }

    // ================= PASS 1: masked max/min =================
    if (wv == 0) tdm_load_tile16x128(centers, lds_offset(&sB[0][0]));
    for (int t = 0; t < NC / 16; ++t) {
        if (wv == 0) __builtin_amdgcn_s_wait_tensorcnt(0);
        __syncthreads();                                        // tile t ready
        if (wv == 0 && t + 1 < NC / 16)                         // overlap next load
            tdm_load_tile16x128(centers + (size_t)(t + 1) * 16 * DIMF,
                                lds_offset(&sB[(t + 1) & 1][0]));

        const float* bptr = &sB[t & 1][lo * DIMF + 2 * hi];     // B mirrors A layout
        v8f acc = {};
        #pragma unroll
        for (int kk = 0; kk < 32; ++kk) {
            v2f b = *(const v2f*)(bptr + kk * 4);
            acc = __builtin_amdgcn_wmma_f32_16x16x4_f32(
                false, av[kk], false, b, (short)0, acc, false, false);
        }
        const int col = t * 16 + lo;
        float yyc = yy[col];
        float wc  = w[col];
        #pragma unroll
        for (int j = 0; j < 8; ++j) {
            float Dv = rxx[j] + yyc - 2.0f * acc[j];
            if (rcid[j] == col) Dv *= wc;                       // own-cluster reweight
            if (Dv > EPS_F) {
                dmax[j] = fmaxf(dmax[j], Dv);
                dmin[j] = fminf(dmin[j], Dv);
            }
        }
        __syncthreads();                                        // done reading buf t
    }

    // Reduce max/min across the 16 column-lanes of each half-wave.
    float base[8];
    #pragma unroll
    for (int j = 0; j < 8; ++j) {
        float mx = dmax[j], mn = dmin[j];
        #pragma unroll
        for (int m = 8; m >= 1; m >>= 1) {
            mx = fmaxf(mx, __shfl_xor(mx, m, 32));
            mn = fminf(mn, __shfl_xor(mn, m, 32));
        }
        base[j] = 0.5f * (mx + mn);
    }

    // ================= PASS 2: masked exp sums (recompute D) =================
    float pos_sum[8] = {0,0,0,0,0,0,0,0};
    float all_sum[8] = {0,0,0,0,0,0,0,0};
    if (wv == 0) tdm_load_tile16x128(centers, lds_offset(&sB[0][0]));
    for (int t = 0; t < NC / 16; ++t) {
        if (wv == 0) __builtin_amdgcn_s_wait_tensorcnt(0);
        __syncthreads();
        if (wv == 0 && t + 1 < NC / 16)
            tdm_load_tile16x128(centers + (size_t)(t + 1) * 16 * DIMF,
                                lds_offset(&sB[(t + 1) & 1][0]));

        const float* bptr = &sB[t & 1][lo * DIMF + 2 * hi];
        v8f acc = {};
        #pragma unroll
        for (int kk = 0; kk < 32; ++kk) {
            v2f b = *(const v2f*)(bptr + kk * 4);
            acc = __builtin_amdgcn_wmma_f32_16x16x4_f32(
                false, av[kk], false, b, (short)0, acc, false, false);
        }
        const int col = t * 16 + lo;
        float yyc = yy[col];
        float wc  = w[col];
        int   cc  = ctg[col];
        #pragma unroll
        for (int j = 0; j < 8; ++j) {
            float Dv = rxx[j] + yyc - 2.0f * acc[j];
            if (rcid[j] == col) Dv *= wc;
            if (Dv > EPS_F) {
                float e = __expf(-ALPHA_F * (Dv - base[j]));
                all_sum[j] += e;
                if (cc == rtgt[j]) pos_sum[j] += e;
            }
        }
        __syncthreads();
    }

    // Final cross-lane reduction + scalar accumulation.
    #pragma unroll
    for (int j = 0; j < 8; ++j) {
        float ps = pos_sum[j], as = all_sum[j];
        #pragma unroll
        for (int m = 8; m >= 1; m >>= 1) {
            ps += __shfl_xor(ps, m, 32);
            as += __shfl_xor(as, m, 32);
        }
        if (lo == 0) {
            float loss = __logf(as) - __logf(ps);   // -log(pos/all)
            unsafeAtomicAdd(out, loss * (1.0f / (float)NS));
        }
    }
}

// ---------------- launcher ----------------
extern "C" void kernel_launch(void* const* d_in, const int* in_sizes, int n_in,
                              void* d_out, int out_size, void* d_ws, size_t ws_size,
                              hipStream_t stream) {
    (void)in_sizes; (void)n_in; (void)out_size; (void)ws_size;

    const float* x   = (const float*)d_in[0];
    const int*   cid = (const int*)d_in[1];
    const int*   tgt = (const int*)d_in[2];
    float*       out = (float*)d_out;

    char* ws = (char*)d_ws;
    float* centers = (float*)(ws);              // 1024*128 f32 = 524288 B
    float* sizes   = (float*)(ws + 524288);     // 1024 f32
    float* yy      = (float*)(ws + 528384);     // 1024 f32
    float* w       = (float*)(ws + 532480);     // 1024 f32
    float* xx      = (float*)(ws + 536576);     // 65536 f32 = 262144 B
    int*   ctg     = (int*)  (ws + 798720);     // 1024 i32

    k_init   <<<dim3(512),              dim3(256), 0, stream>>>(centers, sizes, ctg, out);
    k_accum  <<<dim3(NS * DIMF / 256),  dim3(256), 0, stream>>>(x, cid, tgt, centers, sizes, ctg);
    k_rownorm<<<dim3(NS / 256),         dim3(256), 0, stream>>>(x, xx);
    k_fin    <<<dim3(NC / 4),           dim3(128), 0, stream>>>(centers, sizes, yy, w);
    k_main   <<<dim3(NS / 64),          dim3(128), 0, stream>>>(x, cid, tgt, centers, yy, w, xx, ctg, out);
}